// KernelQLSTM_65481071403675
// MI455X (gfx1250) — compile-verified
//
#include <hip/hip_runtime.h>
#include <hip/hip_bf16.h>

#define GAMMA 0.001f
#define TT 128
#define BB 512
#define II 512
#define NROWS (TT * BB)   // 65536

typedef float v2f __attribute__((ext_vector_type(2)));
typedef float v8f __attribute__((ext_vector_type(8)));

__device__ __forceinline__ float fast_sigmoid(float x) {
    return __builtin_amdgcn_rcpf(1.0f + __expf(-x));
}
__device__ __forceinline__ float fast_tanh(float x) {
    // 1 - 2/(e^{2x}+1); saturates correctly at +-1.
    return 1.0f - 2.0f * __builtin_amdgcn_rcpf(__expf(2.0f * x) + 1.0f);
}

// ---------------------------------------------------------------------------
// Pack kernel. Gate-column index is k = q*4 + fam so the scan's lane q reads
// its 4 per-family projections as one contiguous float4.
//   Wt[512][16]  : x-part weights (column k)
//   Wh[16][4]    : hidden-part weights (row k, col h)
//   biasv[16], thetav[16], kvh[4]
// ---------------------------------------------------------------------------
__global__ void qlstm_pack(const float* __restrict__ Wf, const float* __restrict__ bf,
                           const float* __restrict__ Wi, const float* __restrict__ bi,
                           const float* __restrict__ Wu, const float* __restrict__ bu,
                           const float* __restrict__ Wo, const float* __restrict__ bo,
                           const float* __restrict__ thf, const float* __restrict__ thi,
                           const float* __restrict__ thu, const float* __restrict__ tho,
                           const float* __restrict__ kv,
                           float* __restrict__ Wt, float* __restrict__ Wh,
                           float* __restrict__ biasv, float* __restrict__ thetav,
                           float* __restrict__ kvh) {
    const float* Wfam[4] = {Wf, Wi, Wu, Wo};
    const float* bfam[4] = {bf, bi, bu, bo};
    const float* tfam[4] = {thf, thi, thu, tho};
    int tid = blockIdx.x * blockDim.x + threadIdx.x;
    int nthreads = blockDim.x * gridDim.x;
    // W rows have length D = 516 (512 input + 4 hidden)
    for (int idx = tid; idx < II * 16; idx += nthreads) {
        int d = idx >> 4;
        int k = idx & 15;
        int q = k >> 2, fam = k & 3;          // k = q*4 + fam
        Wt[idx] = Wfam[fam][q * 516 + d];
    }
    if (tid < 16) {
        int q = tid >> 2, fam = tid & 3;
        biasv[tid]  = bfam[fam][q];
        thetav[tid] = tfam[fam][q];
        #pragma unroll
        for (int h = 0; h < 4; ++h)
            Wh[tid * 4 + h] = Wfam[fam][q * 516 + 512 + h];
    }
    if (tid < 4) kvh[tid] = kv[512 + tid];
}

// ---------------------------------------------------------------------------
// Phase 1: P[row][16] = X[row] @ Wt + bias via V_WMMA_F32_16X16X4_F32;
// one 16-row tile per wave, K-loop over 512. Sx[row] = sum_d (x-kv)^2 rides
// along on the same A-fragment loads (pk_fma), reduced with one shfl_xor(16).
// HBM floor: 128 MB input read once ~= 5.5 us @ 23.3 TB/s.
// ---------------------------------------------------------------------------
__global__ void qlstm_gemm(const float* __restrict__ X,     // [NROWS][512]
                           const float* __restrict__ Wt,    // [512][16]
                           const float* __restrict__ biasv, // [16]
                           const float* __restrict__ kv,    // raw kernel_vector [516]
                           float* __restrict__ P,           // [NROWS][16]
                           float* __restrict__ Sx) {        // [NROWS]
    const int lane   = threadIdx.x & 31;
    const int wave   = threadIdx.x >> 5;
    const int wavesB = blockDim.x >> 5;
    const int tile   = blockIdx.x * wavesB + wave;   // 0..4095
    const int row16  = tile << 4;
    const int rA     = lane & 15;          // A row within tile; also B column N
    const int khalf  = (lane >> 4) << 1;   // 0 or 2: this lane's K-pair base
    const int row    = row16 + rA;

    const float* xrow = X + (size_t)row * II;

    v8f acc = {};
    float ssq = 0.0f;

    for (int kk = 0; kk < II; kk += 4) {
        const int k0 = kk + khalf;
        // A fragment: 16x4 f32 (lane = row, vgpr pair = two consecutive K)
        float a0 = xrow[k0];
        float a1 = xrow[k0 + 1];
        float d0 = a0 - kv[k0];
        float d1 = a1 - kv[k0 + 1];
        ssq += d0 * d0 + d1 * d1;
        v2f A; A.x = a0; A.y = a1;
        // B fragment: 4x16 f32 (lane = column N, vgpr pair = two consecutive K)
        v2f Bv;
        Bv.x = Wt[(size_t)k0 * 16 + rA];
        Bv.y = Wt[(size_t)(k0 + 1) * 16 + rA];
        acc = __builtin_amdgcn_wmma_f32_16x16x4_f32(
            /*neg_a=*/false, A, /*neg_b=*/false, Bv,
            /*c_mod=*/(short)0, acc, /*reuse_a=*/false, /*reuse_b=*/false);
    }

    // Lane r holds K%4 in {0,1}, lane r+16 holds {2,3}: combine then store.
    ssq += __shfl_xor(ssq, 16, 32);
    if (lane < 16) Sx[row] = ssq;

    // C/D layout: VGPR r, lanes 0-15 -> M=r, lanes 16-31 -> M=r+8; N = lane%16.
    const float bn = biasv[rA];
    const int mbase = row16 + ((lane >> 4) << 3);
    #pragma unroll
    for (int r = 0; r < 8; ++r) {
        P[(size_t)(mbase + r) * 16 + rA] = acc[r] + bn;
    }
}

// ---------------------------------------------------------------------------
// Phase 2: sequential scan, 4 lanes per batch element (lane q owns wire q).
// The 4-qubit circuit reduces analytically to cosine products:
//   C_w = cos(z_w + theta_w)
//   <Z> = (C1*C2*C3, C0*C1, C0*C1*C2, C0*C1*C2*C3)
// Cross-wire traffic is shfl_xor butterflies within the 4-lane group; the
// wire-subset product per lane uses cndmask inclusion masks (no dynamic
// register indexing: permuted tables wl[fam][j] = Wh[row][q^j] etc.).
// ---------------------------------------------------------------------------
__global__ void qlstm_scan(const float* __restrict__ P,      // [NROWS][16], k = q*4+fam
                           const float* __restrict__ Sx,     // [NROWS]
                           const float* __restrict__ Wh,     // [16][4]
                           const float* __restrict__ thetav, // [16]
                           const float* __restrict__ kvh,    // [4]
                           float* __restrict__ outs,         // [T][B][4]
                           float* __restrict__ hx_out,       // [B][4]
                           float* __restrict__ cx_out) {     // [B][4]
    const int tid = blockIdx.x * blockDim.x + threadIdx.x;
    const int b = tid >> 2;
    const int q = tid & 3;
    if (b >= BB) return;

    // Per-lane permuted tables: index j corresponds to wire (q ^ j).
    float khl[4], wl[4][4], th[4];
    #pragma unroll
    for (int j = 0; j < 4; ++j) khl[j] = kvh[q ^ j];
    #pragma unroll
    for (int fam = 0; fam < 4; ++fam) {
        th[fam] = thetav[q * 4 + fam];
        #pragma unroll
        for (int j = 0; j < 4; ++j)
            wl[fam][j] = Wh[(q * 4 + fam) * 4 + (q ^ j)];
    }
    // Wire-subset inclusion masks for <Z_q> (derived from the CNOT ring):
    //   S(0)={1,2,3} S(1)={0,1} S(2)={0,1,2} S(3)={0,1,2,3};  t_j is wire q^j.
    const bool i0 = (q != 0);
    const bool i1 = (q != 2);
    const bool i2 = (q != 1);
    const bool i3 = (q != 1);

    float hxq = 0.0f, cxq = 0.0f;

    for (int t = 0; t < TT; ++t) {
        const size_t rb = (size_t)t * BB + b;
        const float4 p = *(const float4*)(P + rb * 16 + q * 4);
        const float s = Sx[rb];

        // Share hx across the 4-lane group: s_j = hx at wire (q^j).
        const float h0 = hxq;
        const float h1 = __shfl_xor(hxq, 1, 32);
        const float h2 = __shfl_xor(hxq, 2, 32);
        const float h3 = __shfl_xor(hxq, 3, 32);

        float dif, d2;
        dif = h0 - khl[0]; d2  = dif * dif;
        dif = h1 - khl[1]; d2 += dif * dif;
        dif = h2 - khl[2]; d2 += dif * dif;
        dif = h3 - khl[3]; d2 += dif * dif;
        const float gate = __expf(-GAMMA * (s + d2));

        const float pv[4] = {p.x, p.y, p.z, p.w};
        float C[4];
        #pragma unroll
        for (int fam = 0; fam < 4; ++fam) {
            float dot = wl[fam][0] * h0 + wl[fam][1] * h1 +
                        wl[fam][2] * h2 + wl[fam][3] * h3;
            C[fam] = __cosf((pv[fam] + dot) * gate + th[fam]);
        }

        float e[4];
        #pragma unroll
        for (int fam = 0; fam < 4; ++fam) {
            const float y1 = __shfl_xor(C[fam], 1, 32);
            const float y2 = __shfl_xor(C[fam], 2, 32);
            const float y3 = __shfl_xor(C[fam], 3, 32);
            e[fam] = (i0 ? C[fam] : 1.0f) * (i1 ? y1 : 1.0f) *
                     (i2 ? y2 : 1.0f) * (i3 ? y3 : 1.0f);
        }

        const float f = fast_sigmoid(e[0]);
        const float i = fast_sigmoid(e[1]);
        const float g = fast_tanh(e[2]);
        const float o = fast_sigmoid(e[3]);

        cxq = f * cxq + i * g;
        hxq = o * fast_tanh(cxq);
        outs[rb * 4 + q] = hxq;   // coalesced: 4B per lane, contiguous
    }

    hx_out[b * 4 + q] = hxq;
    cx_out[b * 4 + q] = cxq;
}

// ---------------------------------------------------------------------------
// Launch
// ---------------------------------------------------------------------------
extern "C" void kernel_launch(void* const* d_in, const int* in_sizes, int n_in,
                              void* d_out, int out_size, void* d_ws, size_t ws_size,
                              hipStream_t stream) {
    const float* X   = (const float*)d_in[0];   // [128,512,512]
    const float* Wf  = (const float*)d_in[1];
    const float* bf  = (const float*)d_in[2];
    const float* Wi  = (const float*)d_in[3];
    const float* bi  = (const float*)d_in[4];
    const float* Wu  = (const float*)d_in[5];
    const float* bu  = (const float*)d_in[6];
    const float* Wo  = (const float*)d_in[7];
    const float* bo  = (const float*)d_in[8];
    const float* thf = (const float*)d_in[9];
    const float* thi = (const float*)d_in[10];
    const float* thu = (const float*)d_in[11];
    const float* tho = (const float*)d_in[12];
    const float* kv  = (const float*)d_in[13];  // [516]

    float* out    = (float*)d_out;
    float* outs   = out;                         // [128,512,4]
    float* hx_out = out + (size_t)TT * BB * 4;   // [512,4]
    float* cx_out = hx_out + (size_t)BB * 4;     // [512,4]

    // Workspace layout (floats)
    float* ws     = (float*)d_ws;
    float* Wt     = ws;            // 512*16 = 8192
    float* Wh     = ws + 8192;     // 64
    float* biasv  = ws + 8256;     // 16
    float* thetav = ws + 8272;     // 16
    float* kvh    = ws + 8288;     // 16 (padded)
    float* Sx     = ws + 8304;     // 65536
    float* P      = Sx + NROWS;    // 65536*16 (16B-aligned: 73840*4 bytes)

    qlstm_pack<<<32, 256, 0, stream>>>(Wf, bf, Wi, bi, Wu, bu, Wo, bo,
                                       thf, thi, thu, tho, kv,
                                       Wt, Wh, biasv, thetav, kvh);

    // 4096 tiles of 16 rows; 8 waves per 256-thread block -> 512 blocks.
    qlstm_gemm<<<512, 256, 0, stream>>>(X, Wt, biasv, kv, P, Sx);

    // 4 lanes per batch element -> 2048 threads.
    qlstm_scan<<<8, 256, 0, stream>>>(P, Sx, Wh, thetav, kvh,
                                      outs, hx_out, cx_out);
}